// Dist_wrapper_11733850653007
// MI455X (gfx1250) — compile-verified
//
#include <hip/hip_runtime.h>
#include <math.h>

// CDNA5 / gfx1250: wave32, WMMA 16x16x4 f32.
typedef float v2f __attribute__((ext_vector_type(2)));
typedef float v8f __attribute__((ext_vector_type(8)));

#define GRID_N    128
#define NPIX      (GRID_N * GRID_N)    // 16384
#define NSEG      2047                  // 2048 points -> 2047 segments
#define HALF_SEGS 1024                  // LDS-staged segments per pass (2 passes, 1 pad col)
#define HALF_TILES (HALF_SEGS / 16)     // 64 WMMA column-tiles per pass
#define GAMMA_F   200.0f

// ---- cooperative fill of per-segment params into LDS (all 256 threads) ----
// sQ0[i] = ( vx/2, vy/2, -pjx, -pjy )   (WMMA B operands, x2 of K-replication folded)
// sQ1[i] = ( c1, c2, inv_vnorm, vnorm ) (c1 = -pj.v ; c2 = |pj|^2 or 1e30 if masked/pad)
__device__ __forceinline__ void fill_params(const float* __restrict__ kps,
                                            int is_pred, int base,
                                            float4* __restrict__ sQ0,
                                            float4* __restrict__ sQ1)
{
#pragma unroll
    for (int k = 0; k < HALF_SEGS / 256; ++k) {       // 4 uniform iterations
        const int i   = k * 256 + threadIdx.x;
        const int seg = base + i;
        const int segc = (seg < NSEG - 1) ? seg : (NSEG - 1);
        const float* ki = kps + 3 * segc;
        const float pix = (ki[0] - 0.5f) * 2.0f;
        const float piy = (ki[1] - 0.5f) * 2.0f;
        const float pen = ki[2];
        const float pjx = (ki[3] - 0.5f) * 2.0f;
        const float pjy = (ki[4] - 0.5f) * 2.0f;

        const float vx = pix - pjx, vy = piy - pjy;
        const float vnorm = vx * vx + vy * vy;
        const float inv   = (vnorm > 0.0f) ? (1.0f / vnorm) : 0.0f; // zero-seg => t=0 => d2=|u|^2

        const bool masked = is_pred ? (pen > 0.5f) : (pen != 0.0f);
        const bool live   = (seg < NSEG) && !masked;

        const float c1 = -(pjx * vx + pjy * vy);
        const float c2 = live ? (pjx * pjx + pjy * pjy) : 1.0e30f;

        sQ0[i] = make_float4(0.5f * vx, 0.5f * vy, -pjx, -pjy);
        sQ1[i] = make_float4(c1, c2, inv, vnorm);
    }
}

// ---- accumulate min d2 over one LDS-staged half (64 seg-tiles) ----
__device__ __forceinline__ void accum_half(const float4* __restrict__ sQ0,
                                           const float4* __restrict__ sQ1,
                                           v2f A, const float* __restrict__ p2r,
                                           float* __restrict__ minr)
{
    const int lane = threadIdx.x & 31;
    const int col  = lane & 15;

    for (int t = 0; t < HALF_TILES; ++t) {
        const int idx = t * 16 + col;
        const float4 q0 = sQ0[idx];   // ds_load_b128
        const float4 q1 = sQ1[idx];   // ds_load_b128

        v2f B1; B1.x = q0.x; B1.y = q0.y;     // v/2 (both lane halves identical)
        v2f B2; B2.x = q0.z; B2.y = q0.w;     // -pj
        v8f C1, C2;
#pragma unroll
        for (int r = 0; r < 8; ++r) { C1[r] = q1.x; C2[r] = q1.y; }

        // D1[m,n] = p.v - pj.v = uv ;  D2[m,n] = -2 p.pj + |pj|^2
        v8f D1 = __builtin_amdgcn_wmma_f32_16x16x4_f32(
            false, A, false, B1, (short)0, C1, false, false);
        v8f D2 = __builtin_amdgcn_wmma_f32_16x16x4_f32(
            false, A, false, B2, (short)0, C2, false, false);

        const float inv   = q1.z;
        const float vnorm = q1.w;
#pragma unroll
        for (int r = 0; r < 8; ++r) {
            const float uv = D1[r];
            const float u2 = D2[r] + p2r[r];              // |p - pj|^2
            float tc = uv * inv;
            tc = fminf(fmaxf(tc, 0.0f), 1.0f);            // clamp t to [0,1]
            const float a  = fmaf(tc, vnorm, -2.0f * uv); // t|v|^2 - 2uv
            const float d2 = fmaf(tc, a, u2);             // |u - t v|^2
            minr[r] = fminf(minr[r], d2);
        }
    }
}

__global__ __launch_bounds__(256) void dist_loss_kernel(
    const float* __restrict__ pred, const float* __restrict__ gt,
    float* __restrict__ partial)
{
    __shared__ float4 sQ0[HALF_SEGS];   // 16 KB
    __shared__ float4 sQ1[HALF_SEGS];   // 16 KB
    __shared__ float  red[256];         // 1 KB

    const int waveInBlk = threadIdx.x >> 5;
    const int wv   = blockIdx.x * 8 + waveInBlk;  // 1024 waves, 16 pixels each
    const int lane = threadIdx.x & 31;
    const int col  = lane & 15;
    const int half = lane >> 4;

    // wave's 16 pixels: one horizontal strip, row y, columns [x0, x0+16)
    const int   x0 = (wv & 7) * 16;
    const int   y  = wv >> 3;
    const float s  = 2.0f / 127.0f;
    const float py = (float)y * s - 1.0f;

    // A operand: M = lane%16, identical values in both lane halves (K-replication)
    v2f A;
    A.x = (float)(x0 + col) * s - 1.0f;
    A.y = py;

    // |p_M|^2 for the 8 rows this lane's C/D elements cover (M = r + 8*half)
    float p2r[8];
#pragma unroll
    for (int r = 0; r < 8; ++r) {
        const float pxr = (float)(x0 + r + half * 8) * s - 1.0f;
        p2r[r] = pxr * pxr + py * py;
    }

    float gtd[8], prd[8];

#pragma unroll
    for (int set = 0; set < 2; ++set) {
        const float* kps = (set == 0) ? gt : pred;
        float minr[8];
#pragma unroll
        for (int r = 0; r < 8; ++r) minr[r] = 3.0e38f;

        for (int h = 0; h < 2; ++h) {
            __syncthreads();                         // previous LDS readers done
            fill_params(kps, set, h * HALF_SEGS, sQ0, sQ1);
            __syncthreads();                         // params visible to all waves
            accum_half(sQ0, sQ1, A, p2r, minr);
        }

        // min across the 16 segment columns (lanes 0-15 / 16-31 independently),
        // then the single exp per pixel:  max beta = exp(-gamma * min d2)
#pragma unroll
        for (int r = 0; r < 8; ++r) {
            float m = minr[r];
#pragma unroll
            for (int off = 1; off < 16; off <<= 1)
                m = fminf(m, __shfl_xor(m, off, 16));
            const float e = __expf(-GAMMA_F * m);   // 1e30 path underflows to 0
            if (set == 0) gtd[r] = e; else prd[r] = e;
        }
    }

    float sum = 0.0f;
#pragma unroll
    for (int r = 0; r < 8; ++r) {
        const float d = prd[r] - gtd[r];
        sum = fmaf(d, d, sum);
    }
    // each pixel's value is replicated across 16 lanes: keep only col==0
    // (lane 0 covers pixels 0..7, lane 16 covers pixels 8..15)
    const float contrib = (col == 0) ? sum : 0.0f;

    __syncthreads();
    red[threadIdx.x] = contrib;
    __syncthreads();
#pragma unroll
    for (int step = 128; step > 0; step >>= 1) {
        if (threadIdx.x < step) red[threadIdx.x] += red[threadIdx.x + step];
        __syncthreads();
    }
    if (threadIdx.x == 0) partial[blockIdx.x] = red[0];
}

__global__ __launch_bounds__(128) void finalize_kernel(
    const float* __restrict__ partial, float* __restrict__ out)
{
    __shared__ float red[128];
    red[threadIdx.x] = partial[threadIdx.x];
    __syncthreads();
#pragma unroll
    for (int step = 64; step > 0; step >>= 1) {
        if (threadIdx.x < step) red[threadIdx.x] += red[threadIdx.x + step];
        __syncthreads();
    }
    if (threadIdx.x == 0) out[0] = red[0] * (1.0f / (float)NPIX);
}

extern "C" void kernel_launch(void* const* d_in, const int* in_sizes, int n_in,
                              void* d_out, int out_size, void* d_ws, size_t ws_size,
                              hipStream_t stream)
{
    // setup_inputs order: pred_coords (2048*3 f32), gt_coords (2048*3 f32)
    const float* pred = (const float*)d_in[0];
    const float* gt   = (const float*)d_in[1];
    float* partial = (float*)d_ws;   // 128 floats (512 B), fully overwritten each call
    float* out     = (float*)d_out;  // scalar f32 loss

    dist_loss_kernel<<<dim3(128), dim3(256), 0, stream>>>(pred, gt, partial);
    finalize_kernel<<<dim3(1), dim3(128), 0, stream>>>(partial, out);
}